// PatchletsExtractor_44435731644678
// MI455X (gfx1250) — compile-verified
//
#include <hip/hip_runtime.h>
#include <hip/hip_bf16.h>
#include <math.h>

typedef float v2f __attribute__((ext_vector_type(2)));
typedef float v8f __attribute__((ext_vector_type(8)));

#define BT   64
#define NPTS 2048
#define KNN  16

// d_out layout in floats (tuple concatenated flat, reference return order)
#define O_IDX     0
#define O_DIST    (BT*NPTS*KNN)                    //  2,097,152
#define O_PATCH   (2*BT*NPTS*KNN)                  //  4,194,304
#define O_PPTS    (3*BT*NPTS*KNN)                  //  6,291,456
#define O_FEATS   (O_PPTS + BT*NPTS*KNN*3)         // 12,582,912
#define O_NORM    (O_FEATS + BT*NPTS*KNN*6)        // 25,165,824

#define QB 256   // queries per workgroup
#define CB 64    // db columns staged per chunk
#define TS 68    // LDS tile row stride in floats (16B-aligned rows, low conflict)

// ---------------------------------------------------------------------------
// Kernel 1: brute-force KNN via V_WMMA_F32_16X16X4_F32.
// key(q,b) = -2*q.b + |b|^2  (monotone in d2; |q|^2 added at emit time)
// ---------------------------------------------------------------------------
__global__ __launch_bounds__(256)
void knn_wmma_kernel(const float* __restrict__ pts, float* __restrict__ out) {
    __shared__ __align__(16) float tile[QB * TS];   // 69,632 B: d2-key tiles
    __shared__ __align__(16) float4 dbs[CB];        //  1,024 B: staged db pts + |b|^2

    const int tid  = threadIdx.x;
    const int lane = tid & 31;
    const int wave = tid >> 5;
    const int wg   = blockIdx.x;      // 512 workgroups
    const int f    = wg >> 3;         // frame 0..63
    const int qb   = wg & 7;          // query block 0..7

    const int t  = f & 15;
    const int qf = (t == 0) ? f : f - 1;            // x2: previous frame (t=0 dup)
    const float* __restrict__ Q  = pts + (size_t)qf * NPTS * 3;
    const float* __restrict__ DB = pts + (size_t)f  * NPTS * 3;

    // this thread's own query row (for |q|^2 and output)
    const int myrow = qb * QB + tid;
    const float qx = Q[myrow*3+0], qy = Q[myrow*3+1], qz = Q[myrow*3+2];
    const float nq = qx*qx + qy*qy + qz*qz;

    // A fragments (ISA 16x4 f32 layout): lanes 0-15 carry K0/K1, 16-31 carry K2/K3.
    // A row = (-2qx, -2qy, -2qz, 1)
    const bool hi = lane >= 16;
    v2f afrag[2];
#pragma unroll
    for (int r = 0; r < 2; ++r) {
        const int rt   = wave * 2 + r;                 // row tile (16 rows)
        const int qrow = qb * QB + rt * 16 + (lane & 15);
        const float x = Q[qrow*3+0], y = Q[qrow*3+1], z = Q[qrow*3+2];
        afrag[r].x = hi ? (-2.0f * z) : (-2.0f * x);
        afrag[r].y = hi ? 1.0f        : (-2.0f * y);
    }

    // per-thread sorted top-16 (static indices -> stays in VGPRs)
    float bd[KNN]; int bi[KNN];
#pragma unroll
    for (int j = 0; j < KNN; ++j) { bd[j] = 3.4e38f; bi[j] = 0; }

    for (int ch = 0; ch < NPTS / CB; ++ch) {
        // stage 64 db points (+ precomputed |b|^2) into LDS
        if (tid < CB) {
            const int p = ch * CB + tid;
            const float bx = DB[p*3+0], by = DB[p*3+1], bz = DB[p*3+2];
            dbs[tid] = make_float4(bx, by, bz, bx*bx + by*by + bz*bz);
        }
        __syncthreads();  // db staged AND previous chunk fully consumed

        // each wave: 2 row-tiles x 4 col-tiles = 8 WMMAs per chunk
#pragma unroll
        for (int r = 0; r < 2; ++r) {
            const int rowbase = (wave * 2 + r) * 16;
#pragma unroll
            for (int c = 0; c < 4; ++c) {
                const int colbase = c * 16;
                const float4 dp = dbs[colbase + (lane & 15)];
                v2f bfrag;                       // B col = (bx,by,bz,|b|^2)
                bfrag.x = hi ? dp.z : dp.x;
                bfrag.y = hi ? dp.w : dp.y;
                v8f acc = {};
                acc = __builtin_amdgcn_wmma_f32_16x16x4_f32(
                    /*neg_a=*/false, afrag[r], /*neg_b=*/false, bfrag,
                    /*c_mod=*/(short)0, acc, /*reuse_a=*/false, /*reuse_b=*/false);
                // C/D layout: lane%16 = N, vgpr v -> M = v + 8*(lane>=16)
                const int col = colbase + (lane & 15);
                const int rb  = rowbase + (hi ? 8 : 0);
#pragma unroll
                for (int v = 0; v < 8; ++v)
                    tile[(rb + v) * TS + col] = acc[v];
            }
        }
        __syncthreads();  // tiles visible

        // selection: thread tid scans its own query row, 64 keys
        const float* row = &tile[tid * TS];
#pragma unroll 4
        for (int c = 0; c < CB; c += 4) {
            const float4 v4 = *(const float4*)(row + c);
            const float vv[4] = {v4.x, v4.y, v4.z, v4.w};
#pragma unroll
            for (int u = 0; u < 4; ++u) {
                const float v = vv[u];
                if (v < bd[KNN-1]) {            // strict: stable ties like top_k
                    bd[KNN-1] = v; bi[KNN-1] = ch * CB + c + u;
#pragma unroll
                    for (int s = KNN - 1; s > 0; --s) {
                        if (bd[s] < bd[s-1]) {
                            float tv = bd[s]; bd[s] = bd[s-1]; bd[s-1] = tv;
                            int   ti = bi[s]; bi[s] = bi[s-1]; bi[s-1] = ti;
                        }
                    }
                }
            }
        }
        // next iteration's first __syncthreads orders consume -> overwrite
    }

    // emit: idx (exact float) + dist = sqrt(max(key + |q|^2, 0))
    float* oi = out + O_IDX  + (size_t)(f * NPTS + myrow) * KNN;
    float* od = out + O_DIST + (size_t)(f * NPTS + myrow) * KNN;
#pragma unroll
    for (int j = 0; j < KNN; ++j) {
        oi[j] = (float)bi[j];
        od[j] = sqrtf(fmaxf(bd[j] + nq, 0.0f));
    }
}

// ---------------------------------------------------------------------------
// Kernel 2: patchlet tracking chain. Each query row's chain is independent
// across frames -> one thread per row, loop over 64 frames.
// ---------------------------------------------------------------------------
__global__ __launch_bounds__(256)
void chain_kernel(float* __restrict__ out) {
    const int i = blockIdx.x * blockDim.x + threadIdx.x;
    if (i >= NPTS) return;
    const float* idxs = out + O_IDX;
    float*       pat  = out + O_PATCH;

    int a;
    {
        const float* r0 = idxs + (size_t)i * KNN;
        float*       w0 = pat  + (size_t)i * KNN;
#pragma unroll
        for (int j = 0; j < KNN; ++j) w0[j] = r0[j];
        a = (int)r0[0];
    }
    for (int f = 1; f < BT; ++f) {
        const float* r = idxs + ((size_t)f * NPTS + a) * KNN;
        float*       w = pat  + ((size_t)f * NPTS + i) * KNN;
        const float first = r[0];
#pragma unroll
        for (int j = 0; j < KNN; ++j) w[j] = r[j];
        a = (int)first;
    }
}

// ---------------------------------------------------------------------------
// Kernel 3: gather patchlet points, anchor-normalize, build feats.
// feat == points here, so feats = [points | normalized].
// ---------------------------------------------------------------------------
__global__ __launch_bounds__(256)
void gather_kernel(const float* __restrict__ pts, float* __restrict__ out) {
    const long long gid = (long long)blockIdx.x * blockDim.x + threadIdx.x;
    if (gid >= (long long)BT * NPTS * KNN) return;
    const int i = (int)((gid >> 4) & (NPTS - 1));
    const int f = (int)(gid >> 15);

    const float* pat = out + O_PATCH;
    const int pl = (int)pat[gid];
    const float* P = pts + (size_t)f * NPTS * 3;
    const float px = P[pl*3+0], py = P[pl*3+1], pz = P[pl*3+2];

    const int f0   = (f >> 4) << 4;                         // first frame of batch
    const int aidx = (int)pat[((size_t)f0 * NPTS + i) * KNN];  // neighbor 0
    const float* P0 = pts + (size_t)f0 * NPTS * 3;
    const float ax = P0[aidx*3+0], ay = P0[aidx*3+1], az = P0[aidx*3+2];
    const float nx = px - ax, ny = py - ay, nz = pz - az;

    float* pp = out + O_PPTS  + gid * 3;
    pp[0] = px; pp[1] = py; pp[2] = pz;
    float* ft = out + O_FEATS + gid * 6;
    ft[0] = px; ft[1] = py; ft[2] = pz; ft[3] = nx; ft[4] = ny; ft[5] = nz;
    float* nm = out + O_NORM  + gid * 3;
    nm[0] = nx; nm[1] = ny; nm[2] = nz;
}

extern "C" void kernel_launch(void* const* d_in, const int* in_sizes, int n_in,
                              void* d_out, int out_size, void* d_ws, size_t ws_size,
                              hipStream_t stream) {
    (void)in_sizes; (void)n_in; (void)out_size; (void)d_ws; (void)ws_size;
    const float* pts = (const float*)d_in[0];
    float* out = (float*)d_out;

    knn_wmma_kernel<<<dim3(512), dim3(256), 0, stream>>>(pts, out);
    chain_kernel<<<dim3((NPTS + 255) / 256), dim3(256), 0, stream>>>(out);
    const long long tot = (long long)BT * NPTS * KNN;
    gather_kernel<<<dim3((unsigned)((tot + 255) / 256)), dim3(256), 0, stream>>>(pts, out);
}